// EGNNLayer_30829275251278
// MI455X (gfx1250) — compile-verified
//
#include <hip/hip_runtime.h>
#include <hip/hip_bf16.h>
#include <math.h>

// ---------------------------------------------------------------------------
// EGNN layer, fused, fp32 WMMA (V_WMMA_F32_16X16X4_F32) on gfx1250 (wave32)
// M-blocked: each wave processes 32 rows (2 WMMA A-fragments) so every
// B-fragment load feeds two WMMAs.
// ---------------------------------------------------------------------------

typedef __attribute__((ext_vector_type(2))) float v2f;
typedef __attribute__((ext_vector_type(8))) float v8f;

#define HDIM 128
#define EIN  263          // 2*H + 6 + 1
#define KPAD 264          // EIN padded to multiple of 4
#define MT   32           // rows (edges/nodes) per wave

__device__ __forceinline__ float silu_f(float x) {
    return x / (1.0f + __expf(-x));
}

__device__ __forceinline__ v8f wmma_f32(v2f a, v2f b, v8f c) {
    return __builtin_amdgcn_wmma_f32_16x16x4_f32(
        false, a, false, b, (short)0, c, false, false);
}

// ---------------------------------------------------------------------------
// init: zero m_i accumulator, seed pos_new = pos
// ---------------------------------------------------------------------------
__global__ void egnn_init_kernel(const float* __restrict__ pos,
                                 float* __restrict__ m_i,
                                 float* __restrict__ pos_out,
                                 int N) {
    int i = blockIdx.x * blockDim.x + threadIdx.x;
    int total = N * HDIM;
    if (i < total) m_i[i] = 0.0f;
    if (i < N * 3) pos_out[i] = pos[i];
}

// ---------------------------------------------------------------------------
// edge kernel: one wave (32 lanes) handles a tile of 32 edges.
//   m    = silu(edge_input @ eW1 + eb1)          K=263
//   m_ij = silu(m @ eW2 + eb2)                   K=128  -> atomic scatter to m_i
//   t    = silu(m_ij @ cW1 + cb1)                K=128
//   cw   = t @ cW2 (reduction)                   -> atomic scatter to pos_out
// ---------------------------------------------------------------------------
__global__ __launch_bounds__(32) void egnn_edge_kernel(
    const float* __restrict__ h, const float* __restrict__ pos,
    const int* __restrict__ edge_index, const float* __restrict__ edge_attr,
    const float* __restrict__ eW1, const float* __restrict__ eb1,
    const float* __restrict__ eW2, const float* __restrict__ eb2,
    const float* __restrict__ cW1, const float* __restrict__ cb1,
    const float* __restrict__ cW2,
    float* __restrict__ m_i,       // [N,H] accumulator (workspace)
    float* __restrict__ pos_out,   // [N,3] pre-seeded with pos
    int E)
{
    __shared__ float sA[MT][KPAD];   // edge_input tile; cols [0,128) reused for m
    __shared__ float sM[MT][HDIM];   // m_ij tile
    __shared__ int   sRow[MT];
    __shared__ int   sCol[MT];
    __shared__ float sRel[MT][3];
    __shared__ float sDist[MT];
    __shared__ float sCW[MT];

    const int lane = threadIdx.x;        // 0..31
    const int e0   = blockIdx.x * MT;

    {   // per-edge metadata: every lane owns one edge of the 32-edge tile
        int e = e0 + lane;
        int ec = (e < E) ? e : (E - 1);          // clamp, branchless
        int r = edge_index[ec];
        int c = edge_index[E + ec];
        sRow[lane] = r; sCol[lane] = c;
        float dx = pos[r * 3 + 0] - pos[c * 3 + 0];
        float dy = pos[r * 3 + 1] - pos[c * 3 + 1];
        float dz = pos[r * 3 + 2] - pos[c * 3 + 2];
        sRel[lane][0] = dx; sRel[lane][1] = dy; sRel[lane][2] = dz;
        sDist[lane] = sqrtf(dx * dx + dy * dy + dz * dz);
        sCW[lane] = 0.0f;
    }
    __syncthreads();

    // Build edge_input tile [32][264]; branchless, coalesced in K.
    for (int m = 0; m < MT; ++m) {
        int r = sRow[m];
        int c = sCol[m];
        int ea = ((e0 + m) < E ? (e0 + m) : (E - 1)) * 6;
        float d = sDist[m];
        for (int k = lane; k < KPAD; k += 32) {
            float v;
            if (k < HDIM)              v = h[r * HDIM + k];
            else if (k < 2 * HDIM)     v = h[c * HDIM + (k - HDIM)];
            else if (k < 2 * HDIM + 6) v = edge_attr[ea + (k - 2 * HDIM)];
            else if (k == 262)         v = d;
            else                       v = 0.0f;   // k == 263 pad
            sA[m][k] = v;
        }
    }
    __syncthreads();

    const int mrow  = lane & 15;            // A row / B column within 16-tile
    const int koff  = (lane >> 4) * 2;      // K sub-offset per half-wave
    const int mbase = (lane >> 4) * 8;      // C/D: M base for this half-wave

    // ---------------- layer 1: K = 263 -------------------------------------
    v8f accA[8] = {}, accB[8] = {};
    for (int kk = 0; kk < 65; ++kk) {       // K rows 0..259, guard-free
        int k0 = kk * 4 + koff;
        v2f a0, a1;
        a0.x = sA[mrow][k0];      a0.y = sA[mrow][k0 + 1];
        a1.x = sA[mrow + 16][k0]; a1.y = sA[mrow + 16][k0 + 1];
        v2f b[8];
#pragma unroll
        for (int t = 0; t < 8; ++t) {
            int nc = t * 16 + mrow;
            b[t].x = eW1[k0 * HDIM + nc];
            b[t].y = eW1[(k0 + 1) * HDIM + nc];
        }
#pragma unroll
        for (int t = 0; t < 8; ++t) accA[t] = wmma_f32(a0, b[t], accA[t]);
#pragma unroll
        for (int t = 0; t < 8; ++t) accB[t] = wmma_f32(a1, b[t], accB[t]);
    }
    {   // tail step: K rows 260..262 (+ zero pad row 263), select not branch
        int k0 = 260 + koff;               // 260 or 262, always < 263
        int k1 = k0 + 1;                   // 261 or 263
        int k1c = (k1 < EIN) ? k1 : (EIN - 1);
        v2f a0, a1;
        a0.x = sA[mrow][k0];      a0.y = sA[mrow][k1];      // col 263 zeroed
        a1.x = sA[mrow + 16][k0]; a1.y = sA[mrow + 16][k1];
        v2f b[8];
#pragma unroll
        for (int t = 0; t < 8; ++t) {
            int nc = t * 16 + mrow;
            float by = eW1[k1c * HDIM + nc];
            b[t].x = eW1[k0 * HDIM + nc];
            b[t].y = (k1 < EIN) ? by : 0.0f;   // v_cndmask, no EXEC branch
        }
#pragma unroll
        for (int t = 0; t < 8; ++t) accA[t] = wmma_f32(a0, b[t], accA[t]);
#pragma unroll
        for (int t = 0; t < 8; ++t) accB[t] = wmma_f32(a1, b[t], accB[t]);
    }
    __syncthreads();

    // bias + SiLU -> m, stored into sA[:, 0:128]
#pragma unroll
    for (int t = 0; t < 8; ++t) {
        float bias = eb1[t * 16 + mrow];
#pragma unroll
        for (int i = 0; i < 8; ++i) {
            sA[mbase + i][t * 16 + mrow]      = silu_f(accA[t][i] + bias);
            sA[mbase + i + 16][t * 16 + mrow] = silu_f(accB[t][i] + bias);
        }
    }
    __syncthreads();

    // ---------------- layer 2: K = 128 --------------------------------------
    v8f acc2A[8] = {}, acc2B[8] = {};
    for (int kk = 0; kk < 32; ++kk) {
        int k0 = kk * 4 + koff;
        v2f a0, a1;
        a0.x = sA[mrow][k0];      a0.y = sA[mrow][k0 + 1];
        a1.x = sA[mrow + 16][k0]; a1.y = sA[mrow + 16][k0 + 1];
        v2f b[8];
#pragma unroll
        for (int t = 0; t < 8; ++t) {
            int nc = t * 16 + mrow;
            b[t].x = eW2[k0 * HDIM + nc];
            b[t].y = eW2[(k0 + 1) * HDIM + nc];
        }
#pragma unroll
        for (int t = 0; t < 8; ++t) acc2A[t] = wmma_f32(a0, b[t], acc2A[t]);
#pragma unroll
        for (int t = 0; t < 8; ++t) acc2B[t] = wmma_f32(a1, b[t], acc2B[t]);
    }

    // bias + SiLU -> m_ij; keep tile in sM and scatter-add into m_i[col]
    const bool tileFull = (e0 + MT <= E);
#pragma unroll
    for (int t = 0; t < 8; ++t) {
        float bias = eb2[t * 16 + mrow];
#pragma unroll
        for (int i = 0; i < 8; ++i) {
            int M0 = mbase + i;
            int M1 = M0 + 16;
            float s0 = silu_f(acc2A[t][i] + bias);
            float s1 = silu_f(acc2B[t][i] + bias);
            sM[M0][t * 16 + mrow] = s0;
            sM[M1][t * 16 + mrow] = s1;
            if (tileFull || (e0 + M0 < E))
                atomicAdd(&m_i[sCol[M0] * HDIM + t * 16 + mrow], s0);
            if (tileFull || (e0 + M1 < E))
                atomicAdd(&m_i[sCol[M1] * HDIM + t * 16 + mrow], s1);
        }
    }
    __syncthreads();

    // ---------------- coord MLP: K = 128, then dot with cW2 ------------------
    v8f acc3A[8] = {}, acc3B[8] = {};
    for (int kk = 0; kk < 32; ++kk) {
        int k0 = kk * 4 + koff;
        v2f a0, a1;
        a0.x = sM[mrow][k0];      a0.y = sM[mrow][k0 + 1];
        a1.x = sM[mrow + 16][k0]; a1.y = sM[mrow + 16][k0 + 1];
        v2f b[8];
#pragma unroll
        for (int t = 0; t < 8; ++t) {
            int nc = t * 16 + mrow;
            b[t].x = cW1[k0 * HDIM + nc];
            b[t].y = cW1[(k0 + 1) * HDIM + nc];
        }
#pragma unroll
        for (int t = 0; t < 8; ++t) acc3A[t] = wmma_f32(a0, b[t], acc3A[t]);
#pragma unroll
        for (int t = 0; t < 8; ++t) acc3B[t] = wmma_f32(a1, b[t], acc3B[t]);
    }

    // partial dot-products over N: cw[M] = sum_n silu(.) * cW2[n]
    float partA[8] = {0.f, 0.f, 0.f, 0.f, 0.f, 0.f, 0.f, 0.f};
    float partB[8] = {0.f, 0.f, 0.f, 0.f, 0.f, 0.f, 0.f, 0.f};
#pragma unroll
    for (int t = 0; t < 8; ++t) {
        float bias = cb1[t * 16 + mrow];
        float w2   = cW2[t * 16 + mrow];
#pragma unroll
        for (int i = 0; i < 8; ++i) {
            partA[i] += silu_f(acc3A[t][i] + bias) * w2;
            partB[i] += silu_f(acc3B[t][i] + bias) * w2;
        }
    }
#pragma unroll
    for (int i = 0; i < 8; ++i) {
        atomicAdd(&sCW[mbase + i], partA[i]);        // ds_add_f32 reduction
        atomicAdd(&sCW[mbase + i + 16], partB[i]);
    }
    __syncthreads();

    if ((e0 + lane) < E) {
        float w = sCW[lane] / (sDist[lane] + 1e-8f);
        int c = sCol[lane];
        atomicAdd(&pos_out[c * 3 + 0], w * sRel[lane][0]);
        atomicAdd(&pos_out[c * 3 + 1], w * sRel[lane][1]);
        atomicAdd(&pos_out[c * 3 + 2], w * sRel[lane][2]);
    }
}

// ---------------------------------------------------------------------------
// node kernel: one wave per 32 nodes.
//   u     = silu(concat(h, m_i) @ nW1 + nb1)     K=256
//   h_new = h + (u @ nW2 + nb2)                  K=128
// ---------------------------------------------------------------------------
__global__ __launch_bounds__(32) void egnn_node_kernel(
    const float* __restrict__ h, const float* __restrict__ m_i,
    const float* __restrict__ nW1, const float* __restrict__ nb1,
    const float* __restrict__ nW2, const float* __restrict__ nb2,
    float* __restrict__ h_out, int N)
{
    __shared__ float sA[MT][2 * HDIM];
    __shared__ float sU[MT][HDIM];

    const int lane = threadIdx.x;
    const int n0   = blockIdx.x * MT;

    for (int i = lane; i < MT * 2 * HDIM; i += 32) {
        int m = i >> 8;
        int c = i & 255;
        int node = (n0 + m < N) ? (n0 + m) : 0;
        sA[m][c] = (c < HDIM) ? h[node * HDIM + c]
                              : m_i[node * HDIM + (c - HDIM)];
    }
    __syncthreads();

    const int mrow  = lane & 15;
    const int koff  = (lane >> 4) * 2;
    const int mbase = (lane >> 4) * 8;

    // layer 1: K = 256
    v8f accA[8] = {}, accB[8] = {};
    for (int kk = 0; kk < 64; ++kk) {
        int k0 = kk * 4 + koff;
        v2f a0, a1;
        a0.x = sA[mrow][k0];      a0.y = sA[mrow][k0 + 1];
        a1.x = sA[mrow + 16][k0]; a1.y = sA[mrow + 16][k0 + 1];
        v2f b[8];
#pragma unroll
        for (int t = 0; t < 8; ++t) {
            int nc = t * 16 + mrow;
            b[t].x = nW1[k0 * HDIM + nc];
            b[t].y = nW1[(k0 + 1) * HDIM + nc];
        }
#pragma unroll
        for (int t = 0; t < 8; ++t) accA[t] = wmma_f32(a0, b[t], accA[t]);
#pragma unroll
        for (int t = 0; t < 8; ++t) accB[t] = wmma_f32(a1, b[t], accB[t]);
    }
    __syncthreads();

#pragma unroll
    for (int t = 0; t < 8; ++t) {
        float bias = nb1[t * 16 + mrow];
#pragma unroll
        for (int i = 0; i < 8; ++i) {
            sU[mbase + i][t * 16 + mrow]      = silu_f(accA[t][i] + bias);
            sU[mbase + i + 16][t * 16 + mrow] = silu_f(accB[t][i] + bias);
        }
    }
    __syncthreads();

    // layer 2: K = 128, residual, store
    v8f acc2A[8] = {}, acc2B[8] = {};
    for (int kk = 0; kk < 32; ++kk) {
        int k0 = kk * 4 + koff;
        v2f a0, a1;
        a0.x = sU[mrow][k0];      a0.y = sU[mrow][k0 + 1];
        a1.x = sU[mrow + 16][k0]; a1.y = sU[mrow + 16][k0 + 1];
        v2f b[8];
#pragma unroll
        for (int t = 0; t < 8; ++t) {
            int nc = t * 16 + mrow;
            b[t].x = nW2[k0 * HDIM + nc];
            b[t].y = nW2[(k0 + 1) * HDIM + nc];
        }
#pragma unroll
        for (int t = 0; t < 8; ++t) acc2A[t] = wmma_f32(a0, b[t], acc2A[t]);
#pragma unroll
        for (int t = 0; t < 8; ++t) acc2B[t] = wmma_f32(a1, b[t], acc2B[t]);
    }

    const bool tileFull = (n0 + MT <= N);
#pragma unroll
    for (int t = 0; t < 8; ++t) {
        float bias = nb2[t * 16 + mrow];
#pragma unroll
        for (int i = 0; i < 8; ++i) {
            int M0 = mbase + i;
            int M1 = M0 + 16;
            if (tileFull || (n0 + M0 < N)) {
                int idx = (n0 + M0) * HDIM + t * 16 + mrow;
                h_out[idx] = h[idx] + acc2A[t][i] + bias;
            }
            if (tileFull || (n0 + M1 < N)) {
                int idx = (n0 + M1) * HDIM + t * 16 + mrow;
                h_out[idx] = h[idx] + acc2B[t][i] + bias;
            }
        }
    }
}

// ---------------------------------------------------------------------------
// launch
// ---------------------------------------------------------------------------
extern "C" void kernel_launch(void* const* d_in, const int* in_sizes, int n_in,
                              void* d_out, int out_size, void* d_ws, size_t ws_size,
                              hipStream_t stream) {
    const float* h          = (const float*)d_in[0];
    const float* pos        = (const float*)d_in[1];
    const int*   edge_index = (const int*)d_in[2];
    const float* edge_attr  = (const float*)d_in[3];
    const float* eW1        = (const float*)d_in[4];
    const float* eb1        = (const float*)d_in[5];
    const float* eW2        = (const float*)d_in[6];
    const float* eb2        = (const float*)d_in[7];
    const float* nW1        = (const float*)d_in[8];
    const float* nb1        = (const float*)d_in[9];
    const float* nW2        = (const float*)d_in[10];
    const float* nb2        = (const float*)d_in[11];
    const float* cW1        = (const float*)d_in[12];
    const float* cb1        = (const float*)d_in[13];
    const float* cW2        = (const float*)d_in[14];

    const int N = in_sizes[0] / HDIM;     // 50000
    const int E = in_sizes[3] / 6;        // 800000

    float* h_out   = (float*)d_out;
    float* pos_out = (float*)d_out + (size_t)N * HDIM;
    float* m_i     = (float*)d_ws;        // N*H accumulator

    int initN = N * HDIM;
    egnn_init_kernel<<<(initN + 255) / 256, 256, 0, stream>>>(pos, m_i, pos_out, N);

    egnn_edge_kernel<<<(E + MT - 1) / MT, 32, 0, stream>>>(
        h, pos, edge_index, edge_attr,
        eW1, eb1, eW2, eb2, cW1, cb1, cW2,
        m_i, pos_out, E);

    egnn_node_kernel<<<(N + MT - 1) / MT, 32, 0, stream>>>(
        h, m_i, nW1, nb1, nW2, nb2, h_out, N);
}